// DiscriminatorAugment_59923383713892
// MI455X (gfx1250) — compile-verified
//
#include <hip/hip_runtime.h>
#include <stdint.h>

// Problem constants (match reference)
constexpr int   B_  = 128, C_ = 3, H_ = 256, W_ = 256;
constexpr int   CH_ = 64,  CW_ = 64;          // cutout 0.25 * 256
constexpr int   HW  = H_ * W_;
constexpr float PROBF = 0.9f;
constexpr float BRIF = 0.2f, CONF = 0.2f, SATF = 0.2f;

typedef __attribute__((ext_vector_type(2))) float v2f;
typedef __attribute__((ext_vector_type(4))) float v4f;
typedef __attribute__((ext_vector_type(8))) float v8f;

__device__ inline v4f rev4(v4f v) { return __builtin_shufflevector(v, v, 3, 2, 1, 0); }

// ---------------------------------------------------------------------------
// Kernel 1: per-(b,c) spatial mean. One block per plane (384 blocks).
// Streams the 256 KB plane through LDS with double-buffered
// global_load_async_to_lds_b128 (ASYNCcnt path), then reduces:
//   per-thread v4f accumulate -> per-wave sum via v_wmma_f32_16x16x4_f32
//   (A = {partial, 0}, B = ones  =>  D[m][*] = p_m + p_{m+16}) -> LDS -> mu.
// ---------------------------------------------------------------------------
__global__ __launch_bounds__(256) void aug_means_kernel(const float* __restrict__ img,
                                                        float* __restrict__ mu) {
  __shared__ v4f   stage[2][256];   // 2 x 4 KB double buffer
  __shared__ float wsum[8];

  const int    plane = blockIdx.x;                 // 0..383  = b*3 + c
  const float* src   = img + (size_t)plane * HW;   // scalar (SGPR) base
  const unsigned tid = threadIdx.x;

  const unsigned lds0 = (unsigned)(unsigned long long)&stage[0][tid];
  const unsigned lds1 = (unsigned)(unsigned long long)&stage[1][tid];
  const unsigned goff = tid * 16u;                 // per-lane byte offset

  // Prologue: prefetch steps 0 and 1.
  asm volatile("global_load_async_to_lds_b128 %0, %1, %2"
               :: "v"(lds0), "v"(goff), "s"(src) : "memory");
  asm volatile("global_load_async_to_lds_b128 %0, %1, %2"
               :: "v"(lds1), "v"(goff + 4096u), "s"(src) : "memory");

  v4f acc = {0.f, 0.f, 0.f, 0.f};
  constexpr int NSTEP = HW / (256 * 4);            // 64 steps of 4 KB
#pragma unroll 2
  for (int i = 0; i < NSTEP; ++i) {
    if (i < NSTEP - 1) asm volatile("s_wait_asynccnt 1" ::: "memory");
    else               asm volatile("s_wait_asynccnt 0" ::: "memory");
    acc += stage[i & 1][tid];
    // Ensure the LDS read retired before the async engine may overwrite it.
    asm volatile("s_wait_dscnt 0" ::: "memory");
    if (i + 2 < NSTEP) {
      const unsigned nlds = (i & 1) ? lds1 : lds0;
      asm volatile("global_load_async_to_lds_b128 %0, %1, %2"
                   :: "v"(nlds), "v"(goff + (unsigned)(i + 2) * 4096u), "s"(src)
                   : "memory");
    }
  }

  const float p = acc.x + acc.y + acc.z + acc.w;

  // Wave32 reduction via WMMA: D[m][n] = sum_k A[m][k]*1 = p_m + p_{m+16}.
  v2f a;   a.x = p;   a.y = 0.f;
  v2f one; one.x = 1.f; one.y = 1.f;
  v8f cz = {0.f, 0.f, 0.f, 0.f, 0.f, 0.f, 0.f, 0.f};
  v8f d  = __builtin_amdgcn_wmma_f32_16x16x4_f32(false, a, false, one,
                                                 (short)0, cz, false, false);
  // lanes 0-15 hold rowsums m=0..7 in d[0..7]; lanes 16-31 hold m=8..15.
  float t = d[0] + d[1] + d[2] + d[3] + d[4] + d[5] + d[6] + d[7];
  t += __shfl_xor(t, 16, 32);                      // full wave sum, all lanes

  if ((tid & 31u) == 0u) wsum[tid >> 5] = t;
  __syncthreads();
  if (tid == 0) {
    float s = 0.f;
#pragma unroll
    for (int i = 0; i < 8; ++i) s += wsum[i];
    mu[plane] = s * (1.0f / (float)HW);
  }
}

// ---------------------------------------------------------------------------
// Kernel 2: fold the whole augmentation into per-batch affine coefficients.
//   out[ch] = P*x_f[ch] + Q*mean_ch(x_f) + R[ch], plus flip/cutout params.
// Not-applied images become an identity copy (P=1,Q=0,R=0, no flip, no cut).
// ---------------------------------------------------------------------------
__global__ __launch_bounds__(128) void aug_coeff_kernel(
    const float* __restrict__ apply_u, const float* __restrict__ flip_u,
    const float* __restrict__ bri_u,   const float* __restrict__ con_u,
    const float* __restrict__ sat_u,   const int* __restrict__ top_idx,
    const int* __restrict__ left_idx,  const float* __restrict__ mu,
    float* __restrict__ coef) {
  const int b = blockIdx.x * blockDim.x + threadIdx.x;
  if (b >= B_) return;

  const bool apply = apply_u[b] < PROBF;
  bool flip        = flip_u[b] < 0.5f;
  const float bb = 1.0f - BRIF + 2.0f * BRIF * bri_u[b];
  const float cc = 1.0f - CONF + 2.0f * CONF * con_u[b];
  const float ss = 1.0f - SATF + 2.0f * SATF * sat_u[b];

  const float m0 = mu[b * 3 + 0], m1 = mu[b * 3 + 1], m2 = mu[b * 3 + 2];
  const float mbar = (m0 + m1 + m2) * (1.0f / 3.0f);

  float P  = ss * cc * bb;
  float Q  = (1.0f - ss) * cc * bb;
  const float k1 = ss * (1.0f - cc) * bb;
  const float k2 = (1.0f - ss) * (1.0f - cc) * bb;
  float R0 = k1 * m0 + k2 * mbar;
  float R1 = k1 * m1 + k2 * mbar;
  float R2 = k1 * m2 + k2 * mbar;
  int top = top_idx[b], left = left_idx[b];

  if (!apply) {        // identity path, uniform with the applied path
    P = 1.0f; Q = 0.0f; R0 = R1 = R2 = 0.0f;
    flip = false; top = H_; left = W_;             // cutout off-image
  }

  float* cf = coef + b * 8;
  cf[0] = P;  cf[1] = Q;
  cf[2] = R0; cf[3] = R1; cf[4] = R2;
  cf[5] = flip ? 1.0f : 0.0f;
  cf[6] = (float)top;
  cf[7] = (float)left;
}

// ---------------------------------------------------------------------------
// Kernel 3: streaming apply. One thread = one (b,h,w..w+3) across 3 channels.
// 3x b128 loads (mirrored if flipped), affine, cutout mask, 3x b128 NT stores
// (non-temporal so stores don't evict the L2-resident input mid-kernel).
// ---------------------------------------------------------------------------
__global__ __launch_bounds__(256) void aug_apply_kernel(const float* __restrict__ img,
                                                        const float* __restrict__ coef,
                                                        float* __restrict__ out) {
  constexpr int W4 = W_ / 4;
  const int gid = blockIdx.x * blockDim.x + threadIdx.x;
  const int w4 = gid % W4;
  const int h  = (gid / W4) % H_;
  const int b  = gid / (W4 * H_);
  if (b >= B_) return;

  const float* cf = coef + b * 8;
  const float P = cf[0], Q = cf[1];
  const float R0 = cf[2], R1 = cf[3], R2 = cf[4];
  const int flip = (int)cf[5];
  const int top  = (int)cf[6];
  const int left = (int)cf[7];

  const int w    = w4 * 4;
  const int srcw = flip ? (W_ - 4 - w) : w;
  const size_t base = (size_t)b * (C_ * HW) + (size_t)h * W_ + srcw;

  v4f x0 = __builtin_nontemporal_load((const v4f*)(img + base));
  v4f x1 = __builtin_nontemporal_load((const v4f*)(img + base + HW));
  v4f x2 = __builtin_nontemporal_load((const v4f*)(img + base + 2 * HW));
  if (flip) { x0 = rev4(x0); x1 = rev4(x1); x2 = rev4(x2); }

  const v4f g = (x0 + x1 + x2) * (1.0f / 3.0f);
  v4f r0 = P * x0 + Q * g + R0;
  v4f r1 = P * x1 + Q * g + R1;
  v4f r2 = P * x2 + Q * g + R2;

  if ((unsigned)(h - top) < (unsigned)CH_) {       // cutout row range
    v4f m;
    m.x = ((unsigned)(w + 0 - left) < (unsigned)CW_) ? 0.f : 1.f;
    m.y = ((unsigned)(w + 1 - left) < (unsigned)CW_) ? 0.f : 1.f;
    m.z = ((unsigned)(w + 2 - left) < (unsigned)CW_) ? 0.f : 1.f;
    m.w = ((unsigned)(w + 3 - left) < (unsigned)CW_) ? 0.f : 1.f;
    r0 *= m; r1 *= m; r2 *= m;
  }

  const size_t ob = (size_t)b * (C_ * HW) + (size_t)h * W_ + w;
  __builtin_nontemporal_store(r0, (v4f*)(out + ob));
  __builtin_nontemporal_store(r1, (v4f*)(out + ob + HW));
  __builtin_nontemporal_store(r2, (v4f*)(out + ob + 2 * HW));
}

// ---------------------------------------------------------------------------
extern "C" void kernel_launch(void* const* d_in, const int* in_sizes, int n_in,
                              void* d_out, int out_size, void* d_ws, size_t ws_size,
                              hipStream_t stream) {
  (void)in_sizes; (void)n_in; (void)out_size; (void)ws_size;
  const float* img     = (const float*)d_in[0];
  const float* apply_u = (const float*)d_in[1];
  const float* flip_u  = (const float*)d_in[2];
  const float* bri_u   = (const float*)d_in[3];
  const float* con_u   = (const float*)d_in[4];
  const float* sat_u   = (const float*)d_in[5];
  const int*   top_i   = (const int*)d_in[6];
  const int*   left_i  = (const int*)d_in[7];
  float*       out     = (float*)d_out;

  float* mu   = (float*)d_ws;       // 384 floats: per-(b,c) spatial means
  float* coef = mu + B_ * C_;       // 128 * 8 floats: per-batch affine params

  aug_means_kernel<<<B_ * C_, 256, 0, stream>>>(img, mu);
  aug_coeff_kernel<<<1, 128, 0, stream>>>(apply_u, flip_u, bri_u, con_u, sat_u,
                                          top_i, left_i, mu, coef);
  const int total = B_ * H_ * (W_ / 4);            // 2,097,152 threads
  aug_apply_kernel<<<total / 256, 256, 0, stream>>>(img, coef, out);
}